// GNNModel_46901042873050
// MI455X (gfx1250) — compile-verified
//
#include <hip/hip_runtime.h>
#include <hip/hip_bf16.h>

typedef __attribute__((ext_vector_type(2))) float v2f;
typedef __attribute__((ext_vector_type(8))) float v8f;

// ---------------------------------------------------------------- utilities

__global__ void k_fill(float* __restrict__ p, float v, size_t n) {
    size_t i = (size_t)blockIdx.x * blockDim.x + threadIdx.x;
    if (i < n) p[i] = v;
}

// deg[dst] += 1 for every edge (self-loops pre-seeded via fill with 1.0)
__global__ void k_deg_count(const int* __restrict__ dst, float* __restrict__ deg, int E) {
    int e = blockIdx.x * blockDim.x + threadIdx.x;
    if (e < E) atomicAdd(&deg[dst[e]], 1.0f);
}

// dis = rsqrt(deg)   (deg >= 1 always because of the self-loop)
__global__ void k_rsqrt(float* __restrict__ d, int N) {
    int i = blockIdx.x * blockDim.x + threadIdx.x;
    if (i < N) d[i] = rsqrtf(d[i]);
}

// ---------------------------------------------------------------- GEMM 1
// hs1[N x 32] = (x[N x 256] @ W1[256 x 32]) * dis[row]
// 256 threads = 8 waves; each wave computes a 16-row x 32-col tile.
__global__ __launch_bounds__(256) void k_gemm1(const float* __restrict__ x,
                                               const float* __restrict__ W1,
                                               const float* __restrict__ dis,
                                               float* __restrict__ hs1, int N) {
    __shared__ float sW[256 * 32];  // 32 KB, whole W1
    int tid = threadIdx.x;
    {
        const float4* W4 = (const float4*)W1;
        float4* sW4 = (float4*)sW;
#pragma unroll
        for (int i = 0; i < 8; ++i) sW4[tid + 256 * i] = W4[tid + 256 * i];
    }
    __syncthreads();

    int wave = tid >> 5;
    int lane = tid & 31;
    int rowBase = blockIdx.x * 128 + wave * 16;
    int m  = lane & 15;        // M (for A) / N (for B/C) within tile
    int kh = (lane >> 4) * 2;  // K-pair select per lane half
    int row = rowBase + m;
    int crow = row < N ? row : N - 1;  // clamp for safe addressing; stores are guarded
    const float* xrow = x + (size_t)crow * 256;

    v8f c0 = {}, c1 = {};
    for (int k = 0; k < 256; k += 4) {
        int kk = k + kh;
        v2f a;  a.x = xrow[kk];            a.y = xrow[kk + 1];
        v2f b0; b0.x = sW[kk * 32 + m];    b0.y = sW[(kk + 1) * 32 + m];
        v2f b1; b1.x = sW[kk * 32 + 16 + m]; b1.y = sW[(kk + 1) * 32 + 16 + m];
        c0 = __builtin_amdgcn_wmma_f32_16x16x4_f32(false, a, false, b0, (short)0, c0, false, false);
        c1 = __builtin_amdgcn_wmma_f32_16x16x4_f32(false, a, false, b1, (short)0, c1, false, false);
    }

    int mo = (lane >> 4) * 8;  // C/D layout: M = v + 8*(lane>=16), N = lane&15
#pragma unroll
    for (int v = 0; v < 8; ++v) {
        int r = rowBase + mo + v;
        if (r < N) {
            float d = dis[r];
            hs1[(size_t)r * 32 + m]      = c0[v] * d;
            hs1[(size_t)r * 32 + 16 + m] = c1[v] * d;
        }
    }
}

// ---------------------------------------------------------------- edge agg 1
// agg1[dst] += hs1[src]  (32 features: warp per edge, lane = feature)
__global__ __launch_bounds__(256) void k_agg1(const int* __restrict__ src,
                                              const int* __restrict__ dst,
                                              const float* __restrict__ hs1,
                                              float* __restrict__ agg1, int E) {
    int e = blockIdx.x * 8 + (threadIdx.x >> 5);
    if (e >= E) return;
    int lane = threadIdx.x & 31;
    int s = src[e], d = dst[e];
    atomicAdd(&agg1[(size_t)d * 32 + lane], hs1[(size_t)s * 32 + lane]);
}

// ---------------------------------------------------------------- GEMM 2
// A = relu(dis*(agg1+hs1) + b1)  computed on the fly, K = 32
// hs2[N x 128] = (A @ W2[32 x 128]) * dis[row]
__global__ __launch_bounds__(256) void k_gemm2(const float* __restrict__ agg1,
                                               const float* __restrict__ hs1,
                                               const float* __restrict__ dis,
                                               const float* __restrict__ b1,
                                               const float* __restrict__ W2,
                                               float* __restrict__ hs2, int N) {
    __shared__ float sW[32 * 128];  // 16 KB, whole W2
    __shared__ float sb1[32];
    int tid = threadIdx.x;
    {
        const float4* W4 = (const float4*)W2;
        float4* sW4 = (float4*)sW;
#pragma unroll
        for (int i = 0; i < 4; ++i) sW4[tid + 256 * i] = W4[tid + 256 * i];
    }
    if (tid < 32) sb1[tid] = b1[tid];
    __syncthreads();

    int wave = tid >> 5;
    int lane = tid & 31;
    int rowBase = blockIdx.x * 128 + wave * 16;
    int m  = lane & 15;
    int kh = (lane >> 4) * 2;
    int row = rowBase + m;
    int crow = row < N ? row : N - 1;
    float drow = dis[crow];
    const float* ag = agg1 + (size_t)crow * 32;
    const float* hg = hs1 + (size_t)crow * 32;

    v8f c[8];
#pragma unroll
    for (int t = 0; t < 8; ++t) c[t] = (v8f){};

#pragma unroll
    for (int k = 0; k < 32; k += 4) {
        int kk = k + kh;
        float t0 = drow * (ag[kk] + hg[kk]) + sb1[kk];          // +hg = self-loop term
        float t1 = drow * (ag[kk + 1] + hg[kk + 1]) + sb1[kk + 1];
        v2f a; a.x = t0 > 0.f ? t0 : 0.f; a.y = t1 > 0.f ? t1 : 0.f;  // relu
#pragma unroll
        for (int t = 0; t < 8; ++t) {
            v2f b;
            b.x = sW[kk * 128 + t * 16 + m];
            b.y = sW[(kk + 1) * 128 + t * 16 + m];
            c[t] = __builtin_amdgcn_wmma_f32_16x16x4_f32(false, a, false, b, (short)0, c[t], false, false);
        }
    }

    int mo = (lane >> 4) * 8;
#pragma unroll
    for (int v = 0; v < 8; ++v) {
        int r = rowBase + mo + v;
        if (r < N) {
            float d = dis[r];
#pragma unroll
            for (int t = 0; t < 8; ++t)
                hs2[(size_t)r * 128 + t * 16 + m] = c[t][v] * d;
        }
    }
}

// ---------------------------------------------------------------- edge agg 2
// out[dst] += hs2[src]  (128 features: warp per edge, 4 per lane, coalesced)
__global__ __launch_bounds__(256) void k_agg2(const int* __restrict__ src,
                                              const int* __restrict__ dst,
                                              const float* __restrict__ hs2,
                                              float* __restrict__ out, int E) {
    int e = blockIdx.x * 8 + (threadIdx.x >> 5);
    if (e >= E) return;
    int lane = threadIdx.x & 31;
    int s = src[e], d = dst[e];
    const float* hp = hs2 + (size_t)s * 128;
    float* op = out + (size_t)d * 128;
#pragma unroll
    for (int i = 0; i < 4; ++i)
        atomicAdd(&op[lane + 32 * i], hp[lane + 32 * i]);
}

// ---------------------------------------------------------------- epilogue
// out = dis[i] * (edge_agg + hs2_self) + b2
__global__ void k_final(float* __restrict__ out, const float* __restrict__ hs2,
                        const float* __restrict__ dis, const float* __restrict__ b2,
                        int N) {
    size_t i = (size_t)blockIdx.x * blockDim.x + threadIdx.x;
    size_t total = (size_t)N * 128;
    if (i < total) {
        int node = (int)(i >> 7);
        int f = (int)(i & 127);
        out[i] = dis[node] * (out[i] + hs2[i]) + b2[f];
    }
}

// ---------------------------------------------------------------- launcher

extern "C" void kernel_launch(void* const* d_in, const int* in_sizes, int n_in,
                              void* d_out, int out_size, void* d_ws, size_t ws_size,
                              hipStream_t stream) {
    const float* x  = (const float*)d_in[0];
    const int*   ei = (const int*)d_in[1];
    const float* W1 = (const float*)d_in[2];
    const float* b1 = (const float*)d_in[3];
    const float* W2 = (const float*)d_in[4];
    const float* b2 = (const float*)d_in[5];

    const int N = in_sizes[0] / 256;
    const int E = in_sizes[1] / 2;
    const int* src = ei;
    const int* dst = ei + E;

    // workspace layout (all fits in L2 on MI455X: ~77 MB staged + 51 MB out)
    char* ws = (char*)d_ws;
    size_t off = 0;
    auto align256 = [](size_t v) { return (v + 255) & ~(size_t)255; };
    float* dis  = (float*)(ws + off); off = align256(off + (size_t)N * 4);
    float* hs1  = (float*)(ws + off); off = align256(off + (size_t)N * 32 * 4);
    float* agg1 = (float*)(ws + off); off = align256(off + (size_t)N * 32 * 4);
    float* hs2  = (float*)(ws + off); off = align256(off + (size_t)N * 128 * 4);
    float* out  = (float*)d_out;

    const int T = 256;
    // 1) init accumulators (deg seeded with 1.0 = self-loop)
    k_fill<<<(N + T - 1) / T, T, 0, stream>>>(dis, 1.0f, (size_t)N);
    k_fill<<<((size_t)N * 32 + T - 1) / T, T, 0, stream>>>(agg1, 0.0f, (size_t)N * 32);
    k_fill<<<((size_t)N * 128 + T - 1) / T, T, 0, stream>>>(out, 0.0f, (size_t)N * 128);

    // 2) degrees -> dis = rsqrt(deg)
    k_deg_count<<<(E + T - 1) / T, T, 0, stream>>>(dst, dis, E);
    k_rsqrt<<<(N + T - 1) / T, T, 0, stream>>>(dis, N);

    // 3) layer 1 GEMM (fp32 WMMA) with dis scaling fused
    int gBlocks = (N + 127) / 128;
    k_gemm1<<<gBlocks, T, 0, stream>>>(x, W1, dis, hs1, N);

    // 4) layer 1 edge aggregation
    k_agg1<<<(E + 7) / 8, T, 0, stream>>>(src, dst, hs1, agg1, E);

    // 5) layer 2 GEMM (fp32 WMMA), relu+bias+self-loop fused into A fragment
    k_gemm2<<<gBlocks, T, 0, stream>>>(agg1, hs1, dis, b1, W2, hs2, N);

    // 6) layer 2 edge aggregation directly into d_out
    k_agg2<<<(E + 7) / 8, T, 0, stream>>>(src, dst, hs2, out, E);

    // 7) epilogue: out = dis*(agg + self) + b2
    k_final<<<((size_t)N * 128 + T - 1) / T, T, 0, stream>>>(out, hs2, dis, b2, N);
}